// DacVQ_42631845380238
// MI455X (gfx1250) — compile-verified
//
#include <hip/hip_runtime.h>
#include <hip/hip_bf16.h>

// ---- problem constants ----
#define BB 16
#define TT 4096
#define CC 1024
#define NN 9
#define DC 8
#define KK 1024
#define TOK (BB * TT)                 // 65536
#define Q_SIZE (TOK * CC)             // 67108864
#define IDX_SIZE (TOK * NN)           // 589824

typedef float v2f __attribute__((ext_vector_type(2)));
typedef float v8f __attribute__((ext_vector_type(8)));

// D = A(16x4 f32) * B(4x16 f32) + C(16x16 f32), exact f32 matrix op on CDNA5
#define WMMA_F32(a, b, c) \
  __builtin_amdgcn_wmma_f32_16x16x4_f32(false, (a), false, (b), (short)0, (c), false, false)

// ---- workspace layout (floats) ----
// wsIn : [NN][256 kgroups][32 lanes][2]  = 147456  (in_w^T, WMMA B-operand swizzle)
// wsSim: [NN][64 tiles][2 kblk][32][2]   = 73728   (normalized cb^T, B-operand swizzle)
// wsOut: [NN][64 tiles][2 kblk][32][2]   = 73728   (out_w^T, B-operand swizzle)
#define WS_IN_ELEMS  (NN * 256 * 32 * 2)
#define WS_SIM_ELEMS (NN * 64 * 2 * 32 * 2)
#define WS_OUT_ELEMS (NN * 64 * 2 * 32 * 2)

__global__ void prep_zero_kernel(float* loss) {
  if (blockIdx.x == 0 && threadIdx.x == 0) *loss = 0.0f;
}

// B-operand for in_proj: B[k][d] = in_w[n][d][kc+k], k in 0..3 per WMMA.
// lane<16 holds (B[0][d],B[1][d]) for d=lane; lane>=16 holds (B[2][d],B[3][d]).
__global__ void prep_in_kernel(const float* __restrict__ in_w, float* __restrict__ wsIn) {
  int g = blockIdx.x * blockDim.x + threadIdx.x;
  if (g >= NN * 256 * 32) return;
  int lane = g & 31;
  int kg = (g >> 5) & 255;
  int n = g >> 13;
  int d = lane & 15;
  int kc = kg * 4 + 2 * (lane >> 4);
  float v0 = 0.0f, v1 = 0.0f;
  if (d < DC) {
    const float* p = in_w + ((size_t)(n * DC + d)) * CC + kc;
    v0 = p[0];
    v1 = p[1];
  }
  wsIn[(size_t)g * 2 + 0] = v0;
  wsIn[(size_t)g * 2 + 1] = v1;
}

// B-operand for sim: B[k][code] = normalize(cb[n][code])[kbase+k]
__global__ void prep_sim_kernel(const float* __restrict__ cb, float* __restrict__ wsSim) {
  int g = blockIdx.x * blockDim.x + threadIdx.x;
  if (g >= NN * 64 * 2 * 32) return;
  int lane = g & 31;
  int w = (g >> 5) & 1;
  int t = (g >> 6) & 63;
  int n = g >> 12;
  int code = t * 16 + (lane & 15);
  int kbase = w * 4 + 2 * (lane >> 4);
  const float* row = cb + ((size_t)n * KK + code) * DC;
  float s = 0.0f;
#pragma unroll
  for (int d = 0; d < DC; ++d) s += row[d] * row[d];
  float nrm = fmaxf(sqrtf(s), 1e-12f);
  float inv = 1.0f / nrm;
  wsSim[(size_t)g * 2 + 0] = row[kbase + 0] * inv;
  wsSim[(size_t)g * 2 + 1] = row[kbase + 1] * inv;
}

// B-operand for out_proj: B[k][c] = out_w[n][c][kbase+k]
__global__ void prep_out_kernel(const float* __restrict__ out_w, float* __restrict__ wsOut) {
  int g = blockIdx.x * blockDim.x + threadIdx.x;
  if (g >= NN * 64 * 2 * 32) return;
  int lane = g & 31;
  int w = (g >> 5) & 1;
  int t = (g >> 6) & 63;
  int n = g >> 12;
  int c = t * 16 + (lane & 15);
  int kbase = w * 4 + 2 * (lane >> 4);
  const float* row = out_w + ((size_t)n * CC + c) * DC;
  wsOut[(size_t)g * 2 + 0] = row[kbase + 0];
  wsOut[(size_t)g * 2 + 1] = row[kbase + 1];
}

// LDS budget (floats): res 16384 | zE 128 | zEn 128 | zQ 128 | part 1024 |
//                      amaxV 1024 | amaxI 1024 | bidx 16 | lred 128  => 19984 f = 79936 B
#define LDS_FLOATS 19984

__global__ __launch_bounds__(128) void rvq_main_kernel(
    const float* __restrict__ x, const float* __restrict__ in_b,
    const float* __restrict__ cb, const float* __restrict__ out_b,
    const float* __restrict__ wsIn, const float* __restrict__ wsSim,
    const float* __restrict__ wsOut, float* __restrict__ q_out,
    float* __restrict__ idx_out, float* __restrict__ loss_out) {
  extern __shared__ float smem[];
  float* res = smem;                       // [16][1024]
  float* zE = res + BB * CC;               // [16][8]
  float* zEn = zE + 128;                   // [16][8]
  float* zQ = zEn + 128;                   // [16][8]
  float* part = zQ + 128;                  // [4][8][32]
  float* amaxV = part + 1024;              // [4][8][32]
  int* amaxI = (int*)(amaxV + 1024);       // [4][8][32]
  int* bidx = amaxI + 1024;                // [16]
  float* lred = (float*)(bidx + 16);       // [128]

  const int tid = threadIdx.x;
  const int lane = tid & 31;
  const int wave = tid >> 5;
  const int hi = lane >> 4;   // 0: lanes 0-15, 1: lanes 16-31
  const int lo = lane & 15;
  const long tok0 = (long)blockIdx.x * 16;

  // residual := x tile (16 tokens x 1024 features) resident in LDS for all 9 steps
  for (int i = tid; i < BB * CC; i += 128) res[i] = x[tok0 * CC + i];
  float lacc = 0.0f;
  __syncthreads();

  for (int n = 0; n < NN; ++n) {
    // prefetch next codebook's in_proj weights into cache while we compute
    if (n + 1 < NN)
      __builtin_prefetch(&wsIn[((size_t)(n + 1) * 256 + wave * 64) * 64 + lane * 2], 0, 1);

    // ---------- in_proj: z_e[16x8] = res[16x1024] * in_w^T, K split over 4 waves ----------
    {
      v8f acc = {};
      const int kc0 = wave * 256;
      const float* bp = wsIn + ((size_t)n * 256 + (kc0 >> 2)) * 64 + lane * 2;
      const float* ap = res + lo * CC + 2 * hi;
      for (int kk = 0; kk < 256; kk += 4) {
        v2f a;
        a.x = ap[kc0 + kk + 0];
        a.y = ap[kc0 + kk + 1];
        v2f b;
        b.x = bp[0];
        b.y = bp[1];
        bp += 64;
        acc = WMMA_F32(a, b, acc);
      }
#pragma unroll
      for (int r = 0; r < 8; ++r) part[(wave * 8 + r) * 32 + lane] = acc[r];
    }
    __syncthreads();
    {  // reduce 4 wave-partials; 128 threads = exactly 16 tokens x 8 dims
      int m = tid >> 3, d = tid & 7;
      int rl = (m & 7) * 32 + d + 16 * (m >> 3);
      float v = part[rl] + part[256 + rl] + part[512 + rl] + part[768 + rl];
      zE[m * 8 + d] = v + in_b[n * DC + d];
    }
    __syncthreads();
    if (tid < 16) {  // L2-normalize z_e per token
      float s = 0.0f;
#pragma unroll
      for (int d = 0; d < 8; ++d) { float v = zE[tid * 8 + d]; s += v * v; }
      float inv = 1.0f / fmaxf(sqrtf(s), 1e-12f);
#pragma unroll
      for (int d = 0; d < 8; ++d) zEn[tid * 8 + d] = zE[tid * 8 + d] * inv;
    }
    __syncthreads();

    // ---------- sim + argmax: [16x8] * [8x1024], 64 N-tiles split over waves ----------
    {
      v2f sa0, sa1;
      sa0.x = zEn[lo * 8 + 2 * hi + 0];
      sa0.y = zEn[lo * 8 + 2 * hi + 1];
      sa1.x = zEn[lo * 8 + 4 + 2 * hi + 0];
      sa1.y = zEn[lo * 8 + 4 + 2 * hi + 1];
      float best[8];
      int besti[8];
#pragma unroll
      for (int r = 0; r < 8; ++r) { best[r] = -1e30f; besti[r] = 0x7fffffff; }
      for (int t = wave; t < 64; t += 4) {
        const float* sb = wsSim + ((size_t)n * 64 + t) * 128 + lane * 2;
        v2f b0, b1;
        b0.x = sb[0];  b0.y = sb[1];
        b1.x = sb[64]; b1.y = sb[65];
        v8f s = {};
        s = WMMA_F32(sa0, b0, s);
        s = WMMA_F32(sa1, b1, s);
        int code = t * 16 + lo;
#pragma unroll
        for (int r = 0; r < 8; ++r) {
          float v = s[r];
          if (v > best[r] || (v == best[r] && code < besti[r])) { best[r] = v; besti[r] = code; }
        }
      }
#pragma unroll
      for (int r = 0; r < 8; ++r) {
        amaxV[(wave * 8 + r) * 32 + lane] = best[r];
        amaxI[(wave * 8 + r) * 32 + lane] = besti[r];
      }
    }
    __syncthreads();
    if (tid < 16) {  // final per-token argmax over 4 waves x 16 lanes (JAX first-max tiebreak)
      int m = tid, r = m & 7, lb = 16 * (m >> 3);
      float bv = -1e30f;
      int bi = 0x7fffffff;
      for (int w = 0; w < 4; ++w)
        for (int l = 0; l < 16; ++l) {
          int off = (w * 8 + r) * 32 + lb + l;
          float v = amaxV[off];
          int i = amaxI[off];
          if (v > bv || (v == bv && i < bi)) { bv = v; bi = i; }
        }
      bidx[m] = bi;
      idx_out[(tok0 + m) * NN + n] = (float)bi;  // output indices as (B,T,N) floats
    }
    __syncthreads();
    {  // gather z_q from UNnormalized codebook + loss (1.25 * mse accumulated)
      int m = tid >> 3, d = tid & 7;
      float zq = cb[((size_t)n * KK + bidx[m]) * DC + d];
      zQ[m * 8 + d] = zq;
      float diff = zE[m * 8 + d] - zq;
      lacc += diff * diff;
    }
    __syncthreads();

    // ---------- out_proj: [16x8] * [8x1024]; residual -= (z_q @ out_w^T + b) ----------
    {
      v2f qa0, qa1;
      qa0.x = zQ[lo * 8 + 2 * hi + 0];
      qa0.y = zQ[lo * 8 + 2 * hi + 1];
      qa1.x = zQ[lo * 8 + 4 + 2 * hi + 0];
      qa1.y = zQ[lo * 8 + 4 + 2 * hi + 1];
      for (int t = wave; t < 64; t += 4) {
        const float* ob = wsOut + ((size_t)n * 64 + t) * 128 + lane * 2;
        v2f b0, b1;
        b0.x = ob[0];  b0.y = ob[1];
        b1.x = ob[64]; b1.y = ob[65];
        v8f o = {};
        o = WMMA_F32(qa0, b0, o);
        o = WMMA_F32(qa1, b1, o);
        int c = t * 16 + lo;
        float bias = out_b[n * CC + c];
#pragma unroll
        for (int r = 0; r < 8; ++r) {
          int m = r + 8 * hi;
          res[m * CC + c] -= (o[r] + bias);  // each (m,c) owned by exactly one lane/vgpr
        }
      }
    }
    __syncthreads();
  }

  // q = zq_sum = x - residual_final  (no separate zq_sum storage needed)
  for (int i = tid; i < BB * CC; i += 128)
    q_out[tok0 * CC + i] = x[tok0 * CC + i] - res[i];

  // block-reduce loss, single atomic per workgroup
  lred[tid] = lacc;
  __syncthreads();
  for (int s = 64; s > 0; s >>= 1) {
    if (tid < s) lred[tid] += lred[tid + s];
    __syncthreads();
  }
  if (tid == 0)
    atomicAdd(loss_out, lred[0] * (1.25f / ((float)TOK * (float)DC)));
}

extern "C" void kernel_launch(void* const* d_in, const int* in_sizes, int n_in,
                              void* d_out, int out_size, void* d_ws, size_t ws_size,
                              hipStream_t stream) {
  const float* x = (const float*)d_in[0];
  const float* in_w = (const float*)d_in[1];
  const float* in_b = (const float*)d_in[2];
  const float* cb = (const float*)d_in[3];
  const float* out_w = (const float*)d_in[4];
  const float* out_b = (const float*)d_in[5];

  float* q = (float*)d_out;
  float* idxo = q + Q_SIZE;
  float* loss = q + Q_SIZE + IDX_SIZE;

  float* wsIn = (float*)d_ws;
  float* wsSim = wsIn + WS_IN_ELEMS;
  float* wsOut = wsSim + WS_SIM_ELEMS;

  prep_zero_kernel<<<1, 64, 0, stream>>>(loss);
  prep_in_kernel<<<(NN * 256 * 32 + 255) / 256, 256, 0, stream>>>(in_w, wsIn);
  prep_sim_kernel<<<(NN * 64 * 2 * 32 + 255) / 256, 256, 0, stream>>>(cb, wsSim);
  prep_out_kernel<<<(NN * 64 * 2 * 32 + 255) / 256, 256, 0, stream>>>(out_w, wsOut);

  const size_t lds_bytes = (size_t)LDS_FLOATS * sizeof(float);  // 79936 B (< 320KB WGP LDS)
  (void)hipFuncSetAttribute((const void*)rvq_main_kernel,
                            hipFuncAttributeMaxDynamicSharedMemorySize, (int)lds_bytes);
  rvq_main_kernel<<<TOK / 16, 128, lds_bytes, stream>>>(
      x, in_b, cb, out_b, wsIn, wsSim, wsOut, q, idxo, loss);
}